// projection_CalculateMatrixQ_85813446574644
// MI455X (gfx1250) — compile-verified
//
#include <hip/hip_runtime.h>
#include <stdint.h>

#define B_  32
#define L_  128
#define D_  128
#define BLD (B_ * L_ * D_)     // 524288 floats per intermediate matrix
#define PITCH 132              // 128 + 4 DWORD pad, produced by TDM pad feature

typedef __attribute__((ext_vector_type(2))) float        v2f;
typedef __attribute__((ext_vector_type(8))) float        v8f;
typedef __attribute__((ext_vector_type(4))) unsigned int v4u;
typedef __attribute__((ext_vector_type(4))) int          v4i;
typedef __attribute__((ext_vector_type(8))) int          v8i;

// D = A(16x4) * B(4x16) + C(16x16), all f32, wave32.
__device__ __forceinline__ v8f wmma4(v2f a, v2f b, v8f c) {
    return __builtin_amdgcn_wmma_f32_16x16x4_f32(
        /*neg_a=*/false, a, /*neg_b=*/false, b,
        /*c_mod=*/(short)0, c, /*reuse_a=*/false, /*reuse_b=*/false);
}

// ---------------------------------------------------------------------------
// TDM: DMA a rows x width f32 row-major tile from global into LDS at lds_off.
// pad_enable=1, pad_interval=6 (every 128 DWORDs), pad_amount=3 (4 DWORDs)
// => LDS rows land at a 132-DWORD pitch (kills 16-way bank conflicts on the
//    per-lane column reads feeding WMMA).
// D# bit layout per CDNA5 ISA 8.3/8.4; 2D tensor so remaining groups are zero.
// This toolchain exposes the 6-arg builtin form:
//   (v4u g0, v8i g1, v4i, v4i, v8i, i32 cpol)
// ---------------------------------------------------------------------------
__device__ __forceinline__ void tdm_load_2d_f32(const float* gptr, unsigned lds_off,
                                                unsigned width, unsigned rows,
                                                unsigned row_stride) {
    const unsigned long long ga = (unsigned long long)(uintptr_t)gptr;
    const v4u g0 = { 1u,                                        // count=1 valid user D#
                     lds_off,                                   // lds_addr (bytes)
                     (unsigned)(ga & 0xFFFFFFFFu),              // global_addr[31:0]
                     ((unsigned)(ga >> 32) & 0x01FFFFFFu)       // global_addr[56:32]
                         | 0x80000000u };                       // type=2 (image)
    const v8i g1 = {
        (int)((2u << 16) | (1u << 20) | (6u << 22) | (3u << 25)), // 4B elts, pad 4dw/128dw
        (int)(width << 16),                                     // tensor_dim0[15:0]
        (int)((rows << 16) | (width >> 16)),                    // tdim1[15:0]|tdim0[31:16]
        (int)(((width & 0xFFFFu) << 16) | (rows >> 16)),        // tile_dim0 | tdim1[31:16]
        (int)(rows & 0xFFFFu),                                  // tile_dim1 (tile_dim2=0)
        (int)row_stride,                                        // tensor_dim0_stride[31:0]
        0, 0 };
    const v4i z4 = { 0, 0, 0, 0 };
    const v8i z8 = { 0, 0, 0, 0, 0, 0, 0, 0 };
    __builtin_amdgcn_tensor_load_to_lds(g0, g1, z4, z4, z8, 0);
}

// ---------------------------------------------------------------------------
// Kernel 1: r1,i1,r2,i2 = x @ W.T + bias  (M=4096, N=128, K=128, 4 GEMMs)
// Block = 256 threads (8 waves) handles one (which, 64-row M block).
// Wave 0 TDM-loads W (128x128) and the X block (64x128) into padded LDS.
// Each wave: 4 N-tiles sharing one A fragment per K-step.
// ---------------------------------------------------------------------------
__global__ void proj_gemm_kernel(const float* __restrict__ inR,
                                 const float* __restrict__ inI,
                                 const float* __restrict__ Wr1, const float* __restrict__ br1,
                                 const float* __restrict__ Wi1, const float* __restrict__ bi1,
                                 const float* __restrict__ Wr2, const float* __restrict__ br2,
                                 const float* __restrict__ Wi2, const float* __restrict__ bi2,
                                 float* __restrict__ ws) {
    extern __shared__ float smem[];
    float* Ws = smem;                   // 128 x 132
    float* Xs = smem + D_ * PITCH;      // 64 x 132

    const int which = blockIdx.x & 3;
    const int mBase = (blockIdx.x >> 2) * 64;

    const float* X; const float* W; const float* bias;
    switch (which) {
        case 0:  X = inR; W = Wr1; bias = br1; break;
        case 1:  X = inI; W = Wi1; bias = bi1; break;
        case 2:  X = inR; W = Wr2; bias = br2; break;
        default: X = inI; W = Wi2; bias = bi2; break;
    }
    float* Y = ws + (size_t)which * BLD;

    const int wv   = threadIdx.x >> 5;
    const int lane = threadIdx.x & 31;

    if (wv == 0) {
        tdm_load_2d_f32(W, (unsigned)(uintptr_t)Ws, D_, D_, D_);
        tdm_load_2d_f32(X + (size_t)mBase * D_, (unsigned)(uintptr_t)Xs, D_, 64, D_);
        __builtin_amdgcn_s_wait_tensorcnt(0);
    }
    __syncthreads();

    const int mtl    = wv >> 1;              // local 16-row tile, 0..3
    const int ntBase = (wv & 1) * 4;         // this wave's 4 N-tiles
    const int mrow   = mtl * 16 + (lane & 15);
    const int koff   = (lane >> 4) * 2;
    const int nlo    = lane & 15;

    v8f acc[4] = {};
    for (int k0 = 0; k0 < D_; k0 += 4) {
        const int k = k0 + koff;
        v2f a;
        a.x = Xs[mrow * PITCH + k];
        a.y = Xs[mrow * PITCH + k + 1];
        #pragma unroll
        for (int j = 0; j < 4; ++j) {
            const int n = (ntBase + j) * 16 + nlo;   // B[k][n] = W[n][k]
            v2f b;
            b.x = Ws[n * PITCH + k];
            b.y = Ws[n * PITCH + k + 1];
            acc[j] = wmma4(a, b, acc[j]);
        }
    }

    const int rowBase = mBase + mtl * 16 + ((lane >> 4) << 3);
    #pragma unroll
    for (int j = 0; j < 4; ++j) {
        const int col = (ntBase + j) * 16 + nlo;
        const float bv = bias[col];
        #pragma unroll
        for (int r = 0; r < 8; ++r)
            Y[(size_t)(rowBase + r) * D_ + col] = acc[j][r] + bv;
    }
}

// ---------------------------------------------------------------------------
// Kernel 2: attention logits + softmax over L. Block per batch, thread per l.
// ---------------------------------------------------------------------------
__global__ void att_softmax_kernel(const float* __restrict__ ws,
                                   float* __restrict__ att) {
    __shared__ float red[L_];
    const int b = blockIdx.x;
    const int l = threadIdx.x;

    const float* r1 = ws + (size_t)b * L_ * D_ + (size_t)l * D_;
    const float* i1 = r1 + BLD;
    const float* r2 = r1 + 2 * (size_t)BLD;
    const float* i2 = r1 + 3 * (size_t)BLD;

    float lr = 0.f, li = 0.f;
    for (int d = 0; d < D_; ++d) {
        const float a = r1[d], bb = i1[d], c = r2[d], e = i2[d];
        lr += (a * a - bb * bb) * (c * c - e * e);
        li += (2.f * a * bb) * (2.f * c * e);
    }

    red[l] = lr; __syncthreads();
    for (int s = L_ / 2; s > 0; s >>= 1) {
        if (l < s) red[l] = fmaxf(red[l], red[l + s]);
        __syncthreads();
    }
    const float mR = red[0]; __syncthreads();
    const float eR = __expf(lr - mR);
    red[l] = eR; __syncthreads();
    for (int s = L_ / 2; s > 0; s >>= 1) {
        if (l < s) red[l] += red[l + s];
        __syncthreads();
    }
    const float sumR = red[0]; __syncthreads();
    att[b * L_ + l] = eR / sumR;

    red[l] = li; __syncthreads();
    for (int s = L_ / 2; s > 0; s >>= 1) {
        if (l < s) red[l] = fmaxf(red[l], red[l + s]);
        __syncthreads();
    }
    const float mI = red[0]; __syncthreads();
    const float eI = __expf(li - mI);
    red[l] = eI; __syncthreads();
    for (int s = L_ / 2; s > 0; s >>= 1) {
        if (l < s) red[l] += red[l + s];
        __syncthreads();
    }
    const float sumI = red[0];
    att[B_ * L_ + b * L_ + l] = eI / sumI;
}

// ---------------------------------------------------------------------------
// Kernel 3: Mr[b][d,e] = sum_l attR[l](r2[l,d]r2[l,e] - i2[l,d]i2[l,e])
//           Mi[b][d,e] = sum_l attI[l](i2[l,d]r2[l,e] + r2[l,d]i2[l,e])
// Block = (batch, which). TDM stages r2, i2 (padded pitch); att row in LDS.
// Each wave: one 16-row strip x 8 N-tiles; att scale + sign folded into A.
// ---------------------------------------------------------------------------
__global__ void outer_gemm_kernel(const float* __restrict__ ws,
                                  const float* __restrict__ att,
                                  float* __restrict__ out) {
    extern __shared__ float smem[];
    float* Rs = smem;                        // 128 x 132 : r2[l][d]
    float* Is = smem + L_ * PITCH;           // 128 x 132 : i2[l][d]
    float* As = smem + 2 * L_ * PITCH;       // 128       : att weights

    const int which = blockIdx.x & 1;        // 0 = Mr, 1 = Mi
    const int b     = blockIdx.x >> 1;

    const float* r2 = ws + 2 * (size_t)BLD + (size_t)b * L_ * D_;
    const float* i2 = ws + 3 * (size_t)BLD + (size_t)b * L_ * D_;

    const int wv   = threadIdx.x >> 5;
    const int lane = threadIdx.x & 31;

    if (threadIdx.x < L_)
        As[threadIdx.x] = att[which * (B_ * L_) + b * L_ + threadIdx.x];
    if (wv == 0) {
        tdm_load_2d_f32(r2, (unsigned)(uintptr_t)Rs, D_, L_, D_);
        tdm_load_2d_f32(i2, (unsigned)(uintptr_t)Is, D_, L_, D_);
        __builtin_amdgcn_s_wait_tensorcnt(0);
    }
    __syncthreads();

    const int m    = wv * 16 + (lane & 15);  // output row d (column of staged tiles)
    const int koff = (lane >> 4) * 2;
    const int nlo  = lane & 15;

    v8f acc[8] = {};
    for (int k0 = 0; k0 < L_; k0 += 4) {
        const int k = k0 + koff;             // l index for .x ; .y is k+1
        const float s0  = As[k],                 s1  = As[k + 1];
        const float rm0 = Rs[k * PITCH + m],     rm1 = Rs[(k + 1) * PITCH + m];
        const float im0 = Is[k * PITCH + m],     im1 = Is[(k + 1) * PITCH + m];
        v2f a1, a2;
        if (which == 0) {                    // Mr: attR*r2 , -attR*i2
            a1.x =  s0 * rm0;  a1.y =  s1 * rm1;
            a2.x = -s0 * im0;  a2.y = -s1 * im1;
        } else {                             // Mi: attI*i2 ,  attI*r2
            a1.x =  s0 * im0;  a1.y =  s1 * im1;
            a2.x =  s0 * rm0;  a2.y =  s1 * rm1;
        }
        #pragma unroll
        for (int j = 0; j < 8; ++j) {
            const int n = j * 16 + nlo;
            v2f b1, b2;
            b1.x = Rs[k * PITCH + n];  b1.y = Rs[(k + 1) * PITCH + n];
            b2.x = Is[k * PITCH + n];  b2.y = Is[(k + 1) * PITCH + n];
            acc[j] = wmma4(a1, b1, acc[j]);
            acc[j] = wmma4(a2, b2, acc[j]);
        }
    }

    float* dst = out + (size_t)which * (B_ * D_ * D_) + (size_t)b * D_ * D_;
    const int rowBase = wv * 16 + ((lane >> 4) << 3);
    #pragma unroll
    for (int j = 0; j < 8; ++j) {
        const int col = j * 16 + nlo;
        #pragma unroll
        for (int r = 0; r < 8; ++r)
            dst[(rowBase + r) * D_ + col] = acc[j][r];
    }
}

// ---------------------------------------------------------------------------
extern "C" void kernel_launch(void* const* d_in, const int* in_sizes, int n_in,
                              void* d_out, int out_size, void* d_ws, size_t ws_size,
                              hipStream_t stream) {
    (void)in_sizes; (void)n_in; (void)out_size; (void)ws_size;

    const float* inR = (const float*)d_in[0];
    const float* inI = (const float*)d_in[1];
    const float* Wr1 = (const float*)d_in[2];
    const float* br1 = (const float*)d_in[3];
    const float* Wi1 = (const float*)d_in[4];
    const float* bi1 = (const float*)d_in[5];
    const float* Wr2 = (const float*)d_in[6];
    const float* br2 = (const float*)d_in[7];
    const float* Wi2 = (const float*)d_in[8];
    const float* bi2 = (const float*)d_in[9];

    float* ws  = (float*)d_ws;               // r1 | i1 | r2 | i2 | attR | attI
    float* att = ws + 4 * (size_t)BLD;
    float* out = (float*)d_out;

    // 4 outputs x 64 M-blocks; LDS = (128+64) rows * 132 dw * 4B = 101376 B
    proj_gemm_kernel<<<256, 256, (D_ + 64) * PITCH * sizeof(float), stream>>>(
        inR, inI, Wr1, br1, Wi1, bi1, Wr2, br2, Wi2, bi2, ws);

    att_softmax_kernel<<<B_, L_, 0, stream>>>(ws, att);

    // 32 batches x {Mr,Mi}; LDS = (2*128*132 + 128) * 4 = 135680 B
    outer_gemm_kernel<<<B_ * 2, 256, (2 * L_ * PITCH + L_) * sizeof(float), stream>>>(
        ws, att, out);
}